// CsEstimator_43327630082296
// MI455X (gfx1250) — compile-verified
//
#include <hip/hip_runtime.h>
#include <math.h>
#include <float.h>
#include <stdint.h>

typedef float v2f __attribute__((ext_vector_type(2)));
typedef float v8f __attribute__((ext_vector_type(8)));
typedef unsigned int u32x4 __attribute__((ext_vector_type(4)));
typedef int i32x4 __attribute__((ext_vector_type(4)));
typedef int i32x8 __attribute__((ext_vector_type(8)));

// D = A(16x4,f32) * B(4x16,f32) + C(16x16,f32)
#define WMMA_F32(a, b, c) \
  __builtin_amdgcn_wmma_f32_16x16x4_f32(false, (a), false, (b), (short)0, (c), false, false)

static constexpr int Bn = 16;      // batches
static constexpr int Nn = 15;      // sensors
static constexpr int Tn = 50;      // snapshots
static constexpr int Gg = 4840;    // grid points
static constexpr int G2 = 9680;    // 2*G
static constexpr int K2 = 32;      // 2*N padded to 32 (rows 30,31 zero)
static constexpr int TP = 64;      // T padded
static constexpr int TS = 16;      // WMMA tile dim
static constexpr int SW = 32;      // slab width (2 n-tiles per workgroup)
static constexpr int NSLAB = 2;    // slabs per batch
static constexpr int ITERS = 50;
static constexpr float MUc = 0.01f;

// ---- workspace layout (float units) ----
static constexpr size_t OFF_A    = 0;                         // A[k][g]   [32][9680]
static constexpr size_t OFF_AT   = OFF_A  + (size_t)K2 * G2;  // At[g][k]  [9680][32]
static constexpr size_t OFF_YC   = OFF_AT + (size_t)G2 * K2;  // yc[b][k][t] [16][32][64]
static constexpr size_t OFF_SCAL = OFF_YC + (size_t)Bn * K2 * TP; // [0]=step [1]=thr [2]=step/n
static constexpr size_t OFF_X    = OFF_SCAL + 16;             // x   [16][2][9680][32]
static constexpr size_t OFF_ATY  = OFF_X  + (size_t)Bn * NSLAB * G2 * SW;
static constexpr size_t OFF_NRM  = OFF_ATY + (size_t)Bn * NSLAB * G2 * SW; // [16][64]
static constexpr size_t OFF_MIDX = OFF_NRM + (size_t)Bn * TP; // [16] ints

// -------- build A, At, yc (zero-padded) --------
__global__ void k_setup(const float* __restrict__ yr, const float* __restrict__ yi,
                        const float* __restrict__ Ar, const float* __restrict__ Ai,
                        float* __restrict__ ws) {
  const int NA  = K2 * G2;
  const int NYC = Bn * K2 * TP;
  const int idx = blockIdx.x * blockDim.x + threadIdx.x;
  if (idx < NA) {
    const int k = idx / G2, g = idx % G2;
    float v = 0.0f;
    if (k < Nn)            v = (g < Gg) ? Ar[k * Gg + g] : -Ai[k * Gg + (g - Gg)];
    else if (k < 2 * Nn)   v = (g < Gg) ? Ai[(k - Nn) * Gg + g] : Ar[(k - Nn) * Gg + (g - Gg)];
    ws[OFF_A  + (size_t)k * G2 + g] = v;
    ws[OFF_AT + (size_t)g * K2 + k] = v;
  } else if (idx < NA + NYC) {
    const int j = idx - NA;
    const int b = j / (K2 * TP), k = (j / TP) % K2, t = j % TP;
    float v = 0.0f;
    if (t < Tn) {
      if (k < Nn)          v = yr[(b * Nn + k) * Tn + t];
      else if (k < 2 * Nn) v = yi[(b * Nn + (k - Nn)) * Tn + t];
    }
    ws[OFF_YC + (size_t)j] = v;
  }
}

// -------- Gram = A*A^T via WMMA, power iteration -> step/thr --------
__global__ __launch_bounds__(512) void k_prep(float* __restrict__ ws) {
  __shared__ float red[512 * 16];   // 32 KB
  __shared__ float gram[K2 * K2];
  __shared__ float vv[K2];
  const int tid = threadIdx.x, wave = tid >> 5, lane = tid & 31;
  const int h = lane >> 4, l15 = lane & 15;
  const float* Atp = ws + OFF_AT;

  v8f c00 = {}, c01 = {}, c10 = {}, c11 = {};
  for (int j = wave; j < G2 / 4; j += 16) {
    const int g0 = j * 4;
    v2f a0, a1;
    a0[0] = Atp[(size_t)(g0 + 2 * h + 0) * K2 + l15];
    a0[1] = Atp[(size_t)(g0 + 2 * h + 1) * K2 + l15];
    a1[0] = Atp[(size_t)(g0 + 2 * h + 0) * K2 + 16 + l15];
    a1[1] = Atp[(size_t)(g0 + 2 * h + 1) * K2 + 16 + l15];
    c00 = WMMA_F32(a0, a0, c00);
    c01 = WMMA_F32(a0, a1, c01);
    c10 = WMMA_F32(a1, a0, c10);
    c11 = WMMA_F32(a1, a1, c11);
  }
  for (int half = 0; half < 2; ++half) {
    __syncthreads();
    for (int r = 0; r < 8; ++r) {
      red[(0 * 256 + r * 32 + lane) * 16 + wave] = half ? c10[r] : c00[r];
      red[(1 * 256 + r * 32 + lane) * 16 + wave] = half ? c11[r] : c01[r];
    }
    __syncthreads();
    const int pos = tid;
    float s = 0.0f;
    for (int w = 0; w < 16; ++w) s += red[pos * 16 + w];
    const int tile = pos >> 8, r = (pos >> 5) & 7, ln = pos & 31;
    const int m = half * 16 + r + 8 * (ln >> 4);
    const int n = tile * 16 + (ln & 15);
    gram[m * K2 + n] = s;
  }
  __syncthreads();
  if (tid < 32) {
    vv[lane] = (lane < 30) ? 1.0f : 0.0f;
    for (int it = 0; it < 400; ++it) {
      float w = 0.0f;
      for (int j2 = 0; j2 < K2; ++j2) w += gram[lane * K2 + j2] * vv[j2];
      float ss = w * w;
      for (int off = 16; off > 0; off >>= 1) ss += __shfl_down(ss, off, 32);
      ss = __shfl(ss, 0, 32);
      const float inv = (ss > 0.0f) ? (1.0f / sqrtf(ss)) : 0.0f;
      vv[lane] = w * inv;
    }
    float w = 0.0f;
    for (int j2 = 0; j2 < K2; ++j2) w += gram[lane * K2 + j2] * vv[j2];
    float lp = vv[lane] * w;
    for (int off = 16; off > 0; off >>= 1) lp += __shfl_down(lp, off, 32);
    if (lane == 0) {
      const float lambda = lp;              // sigma_max(A)^2 (v normalized)
      const float step = 30.0f / lambda;    // 1/L, L = lambda/n, n = 2N = 30
      ws[OFF_SCAL + 0] = step;
      ws[OFF_SCAL + 1] = MUc * step;        // thr
      ws[OFF_SCAL + 2] = 1.0f / lambda;     // step/n
    }
  }
}

// -------- persistent ISTA: one workgroup per (batch, 32-col slab) --------
__global__ __launch_bounds__(512) void k_ista(float* __restrict__ ws) {
  __shared__ float red[512 * 16];   // 32 KB cross-wave reduction
  __shared__ float u_lds[K2 * SW];  // u[k][t]       4 KB
  __shared__ float yls[K2 * SW];    // y tile in LDS 4 KB (TDM-staged)
  const int b = blockIdx.x >> 1, tt = blockIdx.x & 1;
  const int tid = threadIdx.x, wave = tid >> 5, lane = tid & 31;
  const int h = lane >> 4, l15 = lane & 15;
  float* xs        = ws + OFF_X   + (size_t)(b * NSLAB + tt) * G2 * SW;
  float* aty       = ws + OFF_ATY + (size_t)(b * NSLAB + tt) * G2 * SW;
  const float* Ap  = ws + OFF_A;
  const float* Atp = ws + OFF_AT;
  const float thr = ws[OFF_SCAL + 1];
  const float sc  = ws[OFF_SCAL + 2];   // step/n

  // ---- TDM: stage y tile [32 rows k][32 cols t] from yc into LDS ----
  if (wave == 0) {
    const unsigned long long ga =
        (unsigned long long)(uintptr_t)(ws + OFF_YC + (size_t)b * K2 * TP + tt * SW);
    const unsigned int la = (unsigned int)(uintptr_t)(&yls[0]);
    u32x4 g0;
    g0[0] = 1u;                                            // count=1 (valid D#)
    g0[1] = la;                                            // lds_addr
    g0[2] = (unsigned int)(ga & 0xFFFFFFFFu);              // global_addr lo
    g0[3] = (unsigned int)((ga >> 32) & 0x1FFFFFFu) | (2u << 30); // addr hi | type=2
    i32x8 g1;
    g1[0] = (int)(2u << 16);                               // data_size=4B, no mask
    g1[1] = (int)((unsigned)TP << 16);                     // tensor_dim0=64 (lo16)
    g1[2] = (int)((unsigned)K2 << 16);                     // tensor_dim1=32 (lo16)
    g1[3] = (int)((unsigned)SW << 16);                     // tile_dim0=32
    g1[4] = (int)K2;                                       // tile_dim1=32
    g1[5] = (int)TP;                                       // tensor_dim0_stride=64
    g1[6] = 0;
    g1[7] = 0;
    i32x4 gz = {0, 0, 0, 0};
#if __clang_major__ >= 23
    i32x8 gz8 = {0, 0, 0, 0, 0, 0, 0, 0};
    __builtin_amdgcn_tensor_load_to_lds(g0, g1, gz, gz, gz8, 0);
#else
    __builtin_amdgcn_tensor_load_to_lds(g0, g1, gz, gz, 0);
#endif
    __builtin_amdgcn_s_wait_tensorcnt(0);
  }
  __syncthreads();

  // prologue: aty = (step/n) * A^T y ; x = 0
  for (int j = wave; j < G2 / TS; j += 16) {
    const int g0r = j * TS;
    v8f acc0 = {}, acc1 = {};
    for (int kk = 0; kk < 8; ++kk) {
      const int k0 = 4 * kk;
      v2f a, b0, b1;
      a[0]  = Ap[(size_t)(k0 + 2 * h + 0) * G2 + g0r + l15];
      a[1]  = Ap[(size_t)(k0 + 2 * h + 1) * G2 + g0r + l15];
      b0[0] = yls[(k0 + 2 * h + 0) * SW + l15];
      b0[1] = yls[(k0 + 2 * h + 1) * SW + l15];
      b1[0] = yls[(k0 + 2 * h + 0) * SW + 16 + l15];
      b1[1] = yls[(k0 + 2 * h + 1) * SW + 16 + l15];
      acc0 = WMMA_F32(a, b0, acc0);
      acc1 = WMMA_F32(a, b1, acc1);
    }
    for (int r = 0; r < 8; ++r) {
      const size_t rowb = (size_t)(g0r + r + 8 * h) * SW;
      aty[rowb + l15]      = sc * acc0[r];
      aty[rowb + 16 + l15] = sc * acc1[r];
      xs[rowb + l15]       = 0.0f;
      xs[rowb + 16 + l15]  = 0.0f;
    }
  }
  __threadfence();
  __syncthreads();

  for (int it = 0; it < ITERS; ++it) {
    // phase A: u = A * x  (2 k-tiles x 2 n-tiles, K-reduce over g)
    v8f a00 = {}, a01 = {}, a10 = {}, a11 = {};
    for (int j = wave; j < G2 / 4; j += 16) {
      const int gg = j * 4;
      v2f p0, p1, b0, b1;
      p0[0] = Atp[(size_t)(gg + 2 * h + 0) * K2 + l15];
      p0[1] = Atp[(size_t)(gg + 2 * h + 1) * K2 + l15];
      p1[0] = Atp[(size_t)(gg + 2 * h + 0) * K2 + 16 + l15];
      p1[1] = Atp[(size_t)(gg + 2 * h + 1) * K2 + 16 + l15];
      b0[0] = xs[(size_t)(gg + 2 * h + 0) * SW + l15];
      b0[1] = xs[(size_t)(gg + 2 * h + 1) * SW + l15];
      b1[0] = xs[(size_t)(gg + 2 * h + 0) * SW + 16 + l15];
      b1[1] = xs[(size_t)(gg + 2 * h + 1) * SW + 16 + l15];
      a00 = WMMA_F32(p0, b0, a00);
      a01 = WMMA_F32(p0, b1, a01);
      a10 = WMMA_F32(p1, b0, a10);
      a11 = WMMA_F32(p1, b1, a11);
    }
    for (int nt = 0; nt < 2; ++nt) {
      for (int r = 0; r < 8; ++r) {
        red[(0 * 256 + r * 32 + lane) * 16 + wave] = nt ? a01[r] : a00[r];
        red[(1 * 256 + r * 32 + lane) * 16 + wave] = nt ? a11[r] : a10[r];
      }
      __syncthreads();
      {
        const int pos = tid;
        float s = 0.0f;
        for (int w = 0; w < 16; ++w) s += red[pos * 16 + w];
        const int tile = pos >> 8, r = (pos >> 5) & 7, ln = pos & 31;
        u_lds[(tile * 16 + r + 8 * (ln >> 4)) * SW + nt * 16 + (ln & 15)] = s;
      }
      __syncthreads();
    }
    // phase B: x = soft(x - (step/n)*A^T u + aty, thr)
    for (int j = wave; j < G2 / TS; j += 16) {
      const int g0r = j * TS;
      v8f acc0 = {}, acc1 = {};
      for (int kk = 0; kk < 8; ++kk) {
        const int k0 = 4 * kk;
        v2f a, b0, b1;
        a[0]  = Ap[(size_t)(k0 + 2 * h + 0) * G2 + g0r + l15];
        a[1]  = Ap[(size_t)(k0 + 2 * h + 1) * G2 + g0r + l15];
        b0[0] = u_lds[(k0 + 2 * h + 0) * SW + l15];
        b0[1] = u_lds[(k0 + 2 * h + 1) * SW + l15];
        b1[0] = u_lds[(k0 + 2 * h + 0) * SW + 16 + l15];
        b1[1] = u_lds[(k0 + 2 * h + 1) * SW + 16 + l15];
        acc0 = WMMA_F32(a, b0, acc0);
        acc1 = WMMA_F32(a, b1, acc1);
      }
      for (int r = 0; r < 8; ++r) {
        const size_t rowb = (size_t)(g0r + r + 8 * h) * SW;
        {
          const size_t gi = rowb + l15;
          const float v  = xs[gi] - sc * acc0[r] + aty[gi];
          const float av = fabsf(v) - thr;
          xs[gi] = (av > 0.0f) ? copysignf(av, v) : 0.0f;
        }
        {
          const size_t gi = rowb + 16 + l15;
          const float v  = xs[gi] - sc * acc1[r] + aty[gi];
          const float av = fabsf(v) - thr;
          xs[gi] = (av > 0.0f) ? copysignf(av, v) : 0.0f;
        }
      }
    }
    __threadfence();
    __syncthreads();
  }
}

// -------- norms[b][t] = sum_g |x_complex| --------
__global__ __launch_bounds__(512) void k_norms(float* __restrict__ ws) {
  const int b = blockIdx.x >> 2, q = blockIdx.x & 3;
  const int tt = q >> 1, coff = (q & 1) * 16;
  const int wave = threadIdx.x >> 5, lane = threadIdx.x & 31;
  const float* xs = ws + OFF_X + (size_t)(b * NSLAB + tt) * G2 * SW;
  const int tg = q * 16 + wave;
  float s = 0.0f;
  for (int g = lane; g < Gg; g += 32) {
    const float xr = xs[(size_t)g * SW + coff + wave];
    const float xi = xs[(size_t)(g + Gg) * SW + coff + wave];
    s += sqrtf(xr * xr + xi * xi + 1e-12f);
  }
  for (int off = 16; off > 0; off >>= 1) s += __shfl_down(s, off, 32);
  if (lane == 0) ws[OFF_NRM + b * TP + tg] = (tg < Tn) ? s : FLT_MAX;
}

__global__ void k_argmin(float* __restrict__ ws) {
  const int b = threadIdx.x;
  if (b < Bn) {
    const float* nr = ws + OFF_NRM + b * TP;
    float mv = nr[0]; int mi = 0;
    for (int t = 1; t < Tn; ++t) { const float v = nr[t]; if (v < mv) { mv = v; mi = t; } }
    ((int*)ws)[OFF_MIDX + b] = mi;
  }
}

// -------- s[b][g] = mean_{b'} mag[b, g, min_idx[b']] --------
__global__ void k_s(const float* __restrict__ ws, float* __restrict__ out) {
  const int b = blockIdx.x / 19;
  const int g = (blockIdx.x % 19) * 256 + threadIdx.x;
  if (g >= Gg) return;
  const int* midx = (const int*)ws + OFF_MIDX;
  float acc = 0.0f;
  for (int bp = 0; bp < Bn; ++bp) {
    const int t = midx[bp];
    const int tt = t >> 5, col = t & 31;
    const float* xs = ws + OFF_X + (size_t)(b * NSLAB + tt) * G2 * SW;
    const float xr = xs[(size_t)g * SW + col];
    const float xi = xs[(size_t)(g + Gg) * SW + col];
    acc += sqrtf(xr * xr + xi * xi + 1e-12f);
  }
  out[64 + b * Gg + g] = acc * (1.0f / Bn);
}

// -------- top-2 per batch -> doa/range --------
__global__ __launch_bounds__(512) void k_topk(const float* __restrict__ ang,
                                              const float* __restrict__ rgs,
                                              float* out) {
  const int b = threadIdx.x >> 5, lane = threadIdx.x & 31;
  const float* s = out + 64 + b * Gg;
  float v1 = -FLT_MAX; int i1 = 0x7fffffff;
  for (int g = lane; g < Gg; g += 32) {
    const float v = s[g];
    if (v > v1 || (v == v1 && g < i1)) { v1 = v; i1 = g; }
  }
  for (int off = 16; off > 0; off >>= 1) {
    const float ov = __shfl_down(v1, off, 32);
    const int   oi = __shfl_down(i1, off, 32);
    if (ov > v1 || (ov == v1 && oi < i1)) { v1 = ov; i1 = oi; }
  }
  v1 = __shfl(v1, 0, 32); i1 = __shfl(i1, 0, 32);
  float v2 = -FLT_MAX; int i2 = 0x7fffffff;
  for (int g = lane; g < Gg; g += 32) {
    if (g == i1) continue;
    const float v = s[g];
    if (v > v2 || (v == v2 && g < i2)) { v2 = v; i2 = g; }
  }
  for (int off = 16; off > 0; off >>= 1) {
    const float ov = __shfl_down(v2, off, 32);
    const int   oi = __shfl_down(i2, off, 32);
    if (ov > v2 || (ov == v2 && oi < i2)) { v2 = ov; i2 = oi; }
  }
  if (lane == 0) {
    out[b * 2 + 0]      = ang[i1 / 40];
    out[b * 2 + 1]      = ang[i2 / 40];
    out[32 + b * 2 + 0] = rgs[i1 % 40];
    out[32 + b * 2 + 1] = rgs[i2 % 40];
  }
}

extern "C" void kernel_launch(void* const* d_in, const int* in_sizes, int n_in,
                              void* d_out, int out_size, void* d_ws, size_t ws_size,
                              hipStream_t stream) {
  (void)in_sizes; (void)n_in; (void)out_size; (void)ws_size;
  const float* yr  = (const float*)d_in[0];
  const float* yi  = (const float*)d_in[1];
  const float* Ar  = (const float*)d_in[2];
  const float* Ai  = (const float*)d_in[3];
  const float* ang = (const float*)d_in[4];
  const float* rgs = (const float*)d_in[5];
  float* ws  = (float*)d_ws;
  float* out = (float*)d_out;

  const int total = K2 * G2 + Bn * K2 * TP;
  k_setup<<<(total + 255) / 256, 256, 0, stream>>>(yr, yi, Ar, Ai, ws);
  k_prep<<<1, 512, 0, stream>>>(ws);
  k_ista<<<Bn * NSLAB, 512, 0, stream>>>(ws);
  k_norms<<<Bn * NSLAB * 2, 512, 0, stream>>>(ws);
  k_argmin<<<1, 32, 0, stream>>>(ws);
  k_s<<<Bn * 19, 256, 0, stream>>>(ws, out);
  k_topk<<<1, 512, 0, stream>>>(ang, rgs, out);
}